// OnlineTripletLoss_15444702397006
// MI455X (gfx1250) — compile-verified
//
#include <hip/hip_runtime.h>
#include <math.h>

typedef __attribute__((ext_vector_type(2))) float v2f;
typedef __attribute__((ext_vector_type(4))) float v4f;
typedef __attribute__((ext_vector_type(8))) float v8f;

#define NROWS 8192
#define DIM   128
#define NTILES (NROWS / 16)
#define TRIPLET_MARGIN 1.0f
#define ROWS_PER_WG 64
#define THREADS 128
#define LDSPAD 132   // 132*4B = 528B row stride: 16B-aligned, +4 banks/row -> conflict-free

// ---------------- kernel 1: squared norms ----------------------------------
__global__ __launch_bounds__(256) void k_sqnorm(const float* __restrict__ x,
                                                float* __restrict__ sq) {
    int i = blockIdx.x * blockDim.x + threadIdx.x;
    if (i < NROWS) {
        const v4f* row = (const v4f*)(x + (size_t)i * DIM);
        float s = 0.f;
#pragma unroll
        for (int j = 0; j < DIM / 4; ++j) {
            v4f v = row[j];
            s += v.x * v.x + v.y * v.y + v.z * v.z + v.w * v.w;
        }
        sq[i] = s;
    }
}

// ---------------- kernel 2: fused WMMA Gram + batch-hard row reductions ----
__global__ __launch_bounds__(THREADS) void k_hardest(const float* __restrict__ x,
                                                     const int*   __restrict__ labels,
                                                     const float* __restrict__ sq,
                                                     float* __restrict__ hp,
                                                     float* __restrict__ hn) {
    __shared__ float lds_xc[2][16][LDSPAD];   // double-buffered 16x128 column strips
    __shared__ float lds_sqc[2][16];
    __shared__ int   lds_labc[2][16];

    const int tid  = threadIdx.x;
    const int wave = tid >> 5;
    const int lane = tid & 31;
    const int half = lane >> 4;       // 0: C rows 0..7, 1: C rows 8..15
    const int ln16 = lane & 15;
    const int koff = half * 2;        // f32 WMMA A/B frag K offset per half

    const int rowStrip = blockIdx.x * ROWS_PER_WG + wave * 16;
    const int mrow     = rowStrip + ln16;

    // Async stage of one 16x128 column strip: 512 x 16B chunks, 4 per thread.
    // global_load_async_to_lds_b128 writes LDS directly (ASYNCcnt-tracked),
    // no VGPR round-trip. In-order completion => s_wait_asynccnt N semantics.
    auto stage = [&](int b, int ct) {
        const int colBase = ct * 16;
#pragma unroll
        for (int it = 0; it < 4; ++it) {
            int f   = tid + it * THREADS;   // 0..511
            int row = f >> 5;
            int c4  = f & 31;
            const float* gsrc = x + (size_t)(colBase + row) * DIM + c4 * 4;
            // low 32 bits of a generic shared pointer = group-segment offset
            unsigned ldsoff = (unsigned)(unsigned long long)(void*)&lds_xc[b][row][c4 * 4];
            asm volatile("global_load_async_to_lds_b128 %0, %1, off"
                         :: "v"(ldsoff), "v"(gsrc) : "memory");
        }
        if (tid < 16) {     // per-tile column norms / labels (tiny, sync path)
            lds_sqc[b][tid]  = sq[colBase + tid];
            lds_labc[b][tid] = labels[colBase + tid];
        }
    };

    // Preload this wave's A-matrix fragments (16x128 strip) into registers:
    // lane needs K in {4k+koff, 4k+koff+1} for k = 0..31  -> 32 x float2
    v2f areg[32];
    {
        const float* rp = x + (size_t)mrow * DIM + koff;
#pragma unroll
        for (int k4 = 0; k4 < 32; ++k4)
            areg[k4] = *(const v2f*)(rp + k4 * 4);
    }

    // Row-side norms / labels for the 8 C-rows this lane's half covers
    float sqR[8];
    int   labR[8];
#pragma unroll
    for (int r = 0; r < 8; ++r) {
        int rr  = rowStrip + half * 8 + r;
        sqR[r]  = sq[rr];
        labR[r] = labels[rr];
    }

    float maxPos[8], minNeg[8];
#pragma unroll
    for (int r = 0; r < 8; ++r) { maxPos[r] = -INFINITY; minNeg[r] = INFINITY; }

    stage(0, 0);   // prime the pipeline

    for (int ct = 0; ct < NTILES; ++ct) {
        const int cur = ct & 1, nxt = cur ^ 1;

        __syncthreads();   // (A) all waves done reading buf[nxt] (prev iter)
        if (ct + 1 < NTILES) {
            stage(nxt, ct + 1);
            // 4 older ops (tile ct) + 4 in flight (tile ct+1): wait for the old 4
            asm volatile("s_wait_asynccnt 0x4" ::: "memory");
        } else {
            asm volatile("s_wait_asynccnt 0x0" ::: "memory");
        }
        __syncthreads();   // (B) every wave's async writes + sq/lab stores visible

        // G_tile = Xr(16x128) * Xc(16x128)^T: two independent WMMA chains
        v8f acc0 = {0.f, 0.f, 0.f, 0.f, 0.f, 0.f, 0.f, 0.f};
        v8f acc1 = {0.f, 0.f, 0.f, 0.f, 0.f, 0.f, 0.f, 0.f};
#pragma unroll
        for (int k4 = 0; k4 < 32; k4 += 2) {
            v2f b0 = *(const v2f*)&lds_xc[cur][ln16][k4 * 4 + koff];
            v2f b1 = *(const v2f*)&lds_xc[cur][ln16][k4 * 4 + 4 + koff];
            acc0 = __builtin_amdgcn_wmma_f32_16x16x4_f32(
                false, areg[k4],     false, b0, (short)0, acc0, false, false);
            acc1 = __builtin_amdgcn_wmma_f32_16x16x4_f32(
                false, areg[k4 + 1], false, b1, (short)0, acc1, false, false);
        }

        // distances + masked running max/min
        const int   colBase = ct * 16;
        const int   col     = colBase + ln16;
        const float sqc     = lds_sqc[cur][ln16];
        const int   labc    = lds_labc[cur][ln16];
#pragma unroll
        for (int r = 0; r < 8; ++r) {
            int   row  = rowStrip + half * 8 + r;
            float g    = acc0[r] + acc1[r];
            float d2   = fmaxf(sqR[r] + sqc - 2.0f * g, 0.0f);
            float dist = sqrtf(d2);
            bool  same = (labR[r] == labc);
            if (same && (row != col)) maxPos[r] = fmaxf(maxPos[r], dist);
            if (!same)                minNeg[r] = fminf(minNeg[r], dist);
        }
    }

    // Reduce across the 16 lanes of each half (C layout: same row per half)
#pragma unroll
    for (int r = 0; r < 8; ++r) {
#pragma unroll
        for (int m = 1; m < 16; m <<= 1) {
            maxPos[r] = fmaxf(maxPos[r], __shfl_xor(maxPos[r], m, 32));
            minNeg[r] = fminf(minNeg[r], __shfl_xor(minNeg[r], m, 32));
        }
    }
    if (ln16 == 0) {  // lanes 0 (rows 0..7) and 16 (rows 8..15) write results
#pragma unroll
        for (int r = 0; r < 8; ++r) {
            int row = rowStrip + half * 8 + r;
            hp[row] = maxPos[r];
            hn[row] = minNeg[r];
        }
    }
}

// ---------------- kernel 3: deterministic final reduction ------------------
__global__ __launch_bounds__(256) void k_finalize(const float* __restrict__ hp,
                                                  const float* __restrict__ hn,
                                                  float* __restrict__ out) {
    __shared__ float ssum[256];
    __shared__ float scnt[256];
    int t = threadIdx.x;
    float s = 0.f, c = 0.f;
    for (int i = t; i < NROWS; i += 256) {
        // hp=-inf (no positive) or hn=+inf (no negative) -> tl = relu(-inf) = 0
        float tl = fmaxf(hp[i] - hn[i] + TRIPLET_MARGIN, 0.0f);
        if (tl > 0.f) { s += tl; c += 1.f; }
    }
    ssum[t] = s; scnt[t] = c;
    __syncthreads();
    for (int off = 128; off > 0; off >>= 1) {
        if (t < off) { ssum[t] += ssum[t + off]; scnt[t] += scnt[t + off]; }
        __syncthreads();
    }
    if (t == 0) out[0] = (scnt[0] > 0.f) ? (ssum[0] / scnt[0]) : 0.f;
}

// ---------------- launcher --------------------------------------------------
extern "C" void kernel_launch(void* const* d_in, const int* in_sizes, int n_in,
                              void* d_out, int out_size, void* d_ws, size_t ws_size,
                              hipStream_t stream) {
    const float* x      = (const float*)d_in[0];
    const int*   labels = (const int*)d_in[1];   // harness: integer -> const int*
    float* wsF = (float*)d_ws;
    float* sq = wsF;                 // [0, 8192)
    float* hp = wsF + NROWS;         // [8192, 16384)
    float* hn = wsF + 2 * NROWS;     // [16384, 24576)

    k_sqnorm  <<<NROWS / 256, 256, 0, stream>>>(x, sq);
    k_hardest <<<NROWS / ROWS_PER_WG, THREADS, 0, stream>>>(x, labels, sq, hp, hn);
    k_finalize<<<1, 256, 0, stream>>>(hp, hn, (float*)d_out);
}